// RelativeAttention_67903432950268
// MI455X (gfx1250) — compile-verified
//
#include <hip/hip_runtime.h>

typedef __attribute__((ext_vector_type(2))) float v2f;
typedef __attribute__((ext_vector_type(4))) float v4f;
typedef __attribute__((ext_vector_type(8))) float v8f;

#define Bc 2
#define Hc 8
#define Sc 512
#define Dc 64
#define QT 16
#define SP 516   // padded LDS row stride (516 % 64 banks = 4 -> staggered rows)

__launch_bounds__(256)
__global__ void relattn_kernel(const float* __restrict__ Q,
                               const float* __restrict__ K,
                               const float* __restrict__ V,
                               const float* __restrict__ R,
                               const int*   __restrict__ M,
                               float* __restrict__ Out,
                               float* __restrict__ P)
{
    __shared__ float sQ[QT * Dc];      // 16x64 Q tile
    __shared__ float sS[QT * SP];      // 16x512 scores / probs

    const int tid  = threadIdx.x;
    const int lane = tid & 31;
    const int wv   = tid >> 5;         // wave id 0..7 (wave32)
    const int wg   = blockIdx.x;       // 0..511
    const int qt   = wg & 31;          // q-tile within (b,h)
    const int bh   = wg >> 5;          // b*H + h
    const int b    = bh >> 3;
    const int q0   = qt * QT;

    const float* Qb = Q + ((size_t)bh * Sc + q0) * Dc;
    const float* Kb = K + (size_t)bh * Sc * Dc;
    const float* Vb = V + (size_t)bh * Sc * Dc;
    const int*   Mb = M + (size_t)b * Sc * Sc;

    // ---- Phase 0: Q tile (1024 floats) -> LDS, one float4 per thread
    {
        v4f t = *reinterpret_cast<const v4f*>(Qb + tid * 4);
        *reinterpret_cast<v4f*>(&sQ[tid * 4]) = t;
    }
    __syncthreads();

    const int mrow = lane & 15;        // M (A-frag) / N (B-frag) index
    const int hi   = lane >> 4;        // lane-half selects K pair per ISA layout

    // ---- Phase 1: scores += Q * K^T via V_WMMA_F32_16X16X4_F32
    // 8 waves x 4 column tiles = all 32 tiles of the 16x512 score block.
    for (int j = 0; j < 4; ++j) {
        const int k0 = (wv * 4 + j) * QT;
        v8f c = {};
        #pragma unroll
        for (int kk = 0; kk < Dc; kk += 4) {
            const int dd = kk + 2 * hi;
            // A: 16x4 of Q  -> lane holds Q[m][dd], Q[m][dd+1]
            v2f a = *reinterpret_cast<const v2f*>(&sQ[mrow * Dc + dd]);
            // B: 4x16 of K^T -> lane holds K[k0+n][dd], K[k0+n][dd+1]
            v2f bb = *reinterpret_cast<const v2f*>(Kb + (size_t)(k0 + mrow) * Dc + dd);
            c = __builtin_amdgcn_wmma_f32_16x16x4_f32(false, a, false, bb,
                                                      (short)0, c, false, false);
        }
        #pragma unroll
        for (int r = 0; r < 8; ++r) {
            const int m = r + 8 * hi;                  // C/D layout: M=r / M=r+8
            sS[m * SP + k0 + mrow] = c[r];
        }
    }
    __syncthreads();

    // ---- Phase 2: QR GEMV stream (1 GiB of R, single use -> NT loads),
    //               scale by 1/sqrt(D), apply mask.
    {
        const int q  = tid >> 4;       // 16 threads per q row
        const int kb = tid & 15;
        v4f qreg[16];
        #pragma unroll
        for (int j = 0; j < 16; ++j)
            qreg[j] = *reinterpret_cast<const v4f*>(&sQ[q * Dc + j * 4]);
        const float* Rq   = R + ((size_t)bh * Sc + q0 + q) * (size_t)Sc * Dc;
        const int*   Mrow = Mb + (size_t)(q0 + q) * Sc;
        for (int i = 0; i < 32; ++i) {
            const int k = kb + 16 * i;
            const v4f* rp = reinterpret_cast<const v4f*>(Rq + (size_t)k * Dc);
            float acc = 0.f;
            #pragma unroll
            for (int j = 0; j < 16; ++j) {
                v4f rv = __builtin_nontemporal_load(rp + j);  // global_load_b128 NT
                acc += qreg[j].x * rv.x + qreg[j].y * rv.y
                     + qreg[j].z * rv.z + qreg[j].w * rv.w;
            }
            float s = (sS[q * SP + k] + acc) * 0.125f;        // 1/sqrt(64)
            if (Mrow[k] == 0) s = -1e9f;
            sS[q * SP + k] = s;
        }
    }
    __syncthreads();

    // ---- Phase 3: softmax over k; 2 rows per wave, wave32 shuffle reductions.
    float* Pb = P + ((size_t)bh * Sc + q0) * Sc;
    #pragma unroll
    for (int rr = 0; rr < 2; ++rr) {
        const int q = wv * 2 + rr;
        float mx = -3.0e38f;
        #pragma unroll
        for (int i = 0; i < 16; ++i)
            mx = fmaxf(mx, sS[q * SP + lane + 32 * i]);
        #pragma unroll
        for (int off = 16; off > 0; off >>= 1)
            mx = fmaxf(mx, __shfl_xor(mx, off, 32));
        float sum = 0.f;
        #pragma unroll
        for (int i = 0; i < 16; ++i) {
            const int k = lane + 32 * i;
            float s = sS[q * SP + k];
            float e = (s < -1.0e8f) ? 0.f : __expf(s - mx);   // masked -> 0
            sS[q * SP + k] = e;
            sum += e;
        }
        #pragma unroll
        for (int off = 16; off > 0; off >>= 1)
            sum += __shfl_xor(sum, off, 32);
        const float inv = (sum > 0.f) ? (1.f / sum) : 0.f;    // all-masked row -> 0
        #pragma unroll
        for (int i = 0; i < 16; ++i) {
            const int k = lane + 32 * i;
            float p = sS[q * SP + k] * inv;
            sS[q * SP + k] = p;
            Pb[(size_t)q * Sc + k] = p;                       // p_attn output
        }
    }
    __syncthreads();

    // ---- Phase 4: Out = P (16x512) x V (512x64) via WMMA fp32; waves 0..3.
    if (wv < 4) {
        const int n0 = wv * QT;
        v8f c = {};
        for (int kk = 0; kk < Sc; kk += 4) {
            const int dd = kk + 2 * hi;
            // A: 16x4 of P -> lane holds P[m][dd], P[m][dd+1] (contiguous in LDS)
            v2f a = *reinterpret_cast<const v2f*>(&sS[mrow * SP + dd]);
            // B: 4x16 of V -> lane holds V[dd][n0+n], V[dd+1][n0+n]
            v2f bb;
            bb.x = Vb[(size_t)dd * Dc + n0 + mrow];
            bb.y = Vb[(size_t)(dd + 1) * Dc + n0 + mrow];
            c = __builtin_amdgcn_wmma_f32_16x16x4_f32(false, a, false, bb,
                                                      (short)0, c, false, false);
        }
        float* Ob = Out + ((size_t)bh * Sc + q0) * Dc;
        #pragma unroll
        for (int r = 0; r < 8; ++r) {
            const int m = r + 8 * hi;
            Ob[(size_t)m * Dc + n0 + mrow] = c[r];
        }
    }
}

extern "C" void kernel_launch(void* const* d_in, const int* in_sizes, int n_in,
                              void* d_out, int out_size, void* d_ws, size_t ws_size,
                              hipStream_t stream) {
    const float* Q = (const float*)d_in[0];
    const float* K = (const float*)d_in[1];
    const float* V = (const float*)d_in[2];
    const float* R = (const float*)d_in[3];
    const int*   M = (const int*)d_in[4];
    float* Out = (float*)d_out;
    float* P   = Out + (size_t)Bc * Hc * Sc * Dc;   // tuple: (out, p_attn) flat

    dim3 grid(Bc * Hc * (Sc / QT));   // 512 workgroups: one per (b,h,q-tile)
    dim3 block(256);                  // 8 wave32 waves
    hipLaunchKernelGGL(relattn_kernel, grid, block, 0, stream, Q, K, V, R, M, Out, P);
}